// HyperCE_58660663329068
// MI455X (gfx1250) — compile-verified
//
#include <hip/hip_runtime.h>
#include <math.h>

typedef __attribute__((ext_vector_type(2))) float v2f;
typedef __attribute__((ext_vector_type(8))) float v8f;

#define DD     128
#define NHEAD  8
#define SPLITN 50000
#define NXR    150000
#define N1R    50000
#define N2R    30000
#define N3R    10000
#define E0N    500000
#define E1N    300000
#define E2N    100000
#define NTYPE  65

// ---------------------------------------------------------------- fill
__global__ void fill_f32(float* __restrict__ p, float v, int n) {
    int i = blockIdx.x * blockDim.x + threadIdx.x;
    if (i < n) p[i] = v;
}

// ---------------------------------------------------------------- GEMM: Y[R,128] = X[R,128] @ W[128,128]
// One wave32 computes a 16x128 strip via V_WMMA_F32_16X16X4_F32.
__global__ __launch_bounds__(256)
void gemm128_wmma(const float* __restrict__ X, const float* __restrict__ W,
                  float* __restrict__ Y, int R) {
    const int wave = threadIdx.x >> 5;
    const int lane = threadIdx.x & 31;
    const int rowBase = blockIdx.x * 128 + wave * 16;
    if (rowBase >= R) return;               // wave-uniform: EXEC stays all-ones for WMMA

    const int m  = lane & 15;               // M (A) / N (B,C) within tile
    const int hi = lane >> 4;               // half-wave select
    const int kh = hi * 2;                  // A/B K sub-offset: lanes16-31 hold K=2,3

    const int arow = (rowBase + m < R) ? (rowBase + m) : (R - 1);
    const float* __restrict__ xrow = X + (size_t)arow * DD;

    v8f c[8] = {};
    #pragma unroll
    for (int k0 = 0; k0 < DD; k0 += 4) {
        v2f a;
        a.x = xrow[k0 + kh];
        a.y = xrow[k0 + kh + 1];
        #pragma unroll
        for (int t = 0; t < 8; ++t) {
            v2f b;
            b.x = W[(size_t)(k0 + kh)     * DD + t * 16 + m];
            b.y = W[(size_t)(k0 + kh + 1) * DD + t * 16 + m];
            c[t] = __builtin_amdgcn_wmma_f32_16x16x4_f32(
                       false, a, false, b, (short)0, c[t], false, false);
        }
    }
    // C/D layout: VGPR i -> row M = i + 8*hi, col N = m
    #pragma unroll
    for (int t = 0; t < 8; ++t) {
        #pragma unroll
        for (int i = 0; i < 8; ++i) {
            int rr = rowBase + i + hi * 8;
            if (rr < R) Y[(size_t)rr * DD + t * 16 + m] = c[t][i];
        }
    }
}

// ---------------------------------------------------------------- float atomic max (ordered-int trick)
__device__ __forceinline__ void atomicMaxF(float* addr, float v) {
    if (v >= 0.0f) atomicMax((int*)addr, __float_as_int(v));
    else           atomicMin((unsigned int*)addr, __float_as_uint(v));
}

// ---------------------------------------------------------------- pass 1: logits + segment max
// One wave32 per edge. Lane l owns cols [4l, 4l+4); head h = l>>2 (16 cols = 4 lanes).
// q/k/ek row gathers are single coalesced b128 loads per lane (512B per wave).
__global__ __launch_bounds__(256)
void edge_logits(const float* __restrict__ q, const float* __restrict__ k,
                 const int* __restrict__ src, const int* __restrict__ dst,
                 const float* __restrict__ ekt, const int* __restrict__ etype,
                 float* __restrict__ logits, float* __restrict__ mbuf,
                 int E, int split) {
    int e = (blockIdx.x * blockDim.x + threadIdx.x) >> 5;
    if (e >= E) return;                     // wave-uniform
    int lane = threadIdx.x & 31;
    int s = src[e], d = dst[e];

    float4 kv = make_float4(0.f, 0.f, 0.f, 0.f);
    if (s >= split) {
        kv = *(const float4*)(k + (size_t)(s - split) * DD + lane * 4);
    }
    if (ekt) {
        float4 ev = *(const float4*)(ekt + (size_t)etype[e] * DD + lane * 4);
        kv.x += ev.x; kv.y += ev.y; kv.z += ev.z; kv.w += ev.w;
    }
    float4 qv = *(const float4*)(q + (size_t)d * DD + lane * 4);

    float part = qv.x * kv.x + qv.y * kv.y + qv.z * kv.z + qv.w * kv.w;
    part += __shfl_xor(part, 1, 32);        // reduce across the 4 lanes of a head
    part += __shfl_xor(part, 2, 32);

    if ((lane & 3) == 0) {
        int h = lane >> 2;
        float lg = part * 0.25f;            // 1/sqrt(16)
        lg = (lg >= 0.0f) ? lg : 0.2f * lg; // leaky relu
        logits[(size_t)e * NHEAD + h] = lg;
        atomicMaxF(&mbuf[(size_t)d * NHEAD + h], lg);
    }
}

// ---------------------------------------------------------------- pass 2: a = exp(l-m); ssum += a; num += a*v
// One wave32 per edge, same lane->column mapping; v/ev gathers coalesced b128.
__global__ __launch_bounds__(256)
void edge_accum(const float* __restrict__ logits, const float* __restrict__ mbuf,
                const float* __restrict__ v,
                const int* __restrict__ src, const int* __restrict__ dst,
                const float* __restrict__ evt, const int* __restrict__ etype,
                float* __restrict__ ssum, float* __restrict__ num,
                int E, int split) {
    int e = (blockIdx.x * blockDim.x + threadIdx.x) >> 5;
    if (e >= E) return;
    int lane = threadIdx.x & 31;
    int h = lane >> 2;
    int s = src[e], d = dst[e];

    float a = __expf(logits[(size_t)e * NHEAD + h] - mbuf[(size_t)d * NHEAD + h]);
    if ((lane & 3) == 0) atomicAdd(&ssum[(size_t)d * NHEAD + h], a);

    float4 vv = make_float4(0.f, 0.f, 0.f, 0.f);
    bool have = false;
    if (s >= split) {
        vv = *(const float4*)(v + (size_t)(s - split) * DD + lane * 4);
        have = true;
    }
    if (evt) {
        float4 ev = *(const float4*)(evt + (size_t)etype[e] * DD + lane * 4);
        vv.x += ev.x; vv.y += ev.y; vv.z += ev.z; vv.w += ev.w;
        have = true;
    }
    if (have) {
        float* np = num + (size_t)d * DD + lane * 4;
        atomicAdd(np + 0, a * vv.x);
        atomicAdd(np + 1, a * vv.y);
        atomicAdd(np + 2, a * vv.z);
        atomicAdd(np + 3, a * vv.w);
    }
}

// ---------------------------------------------------------------- out = beta*s + (1-beta)*num/(ssum+eps)
__global__ void finalize_k(const float* __restrict__ s, const float* __restrict__ num,
                           const float* __restrict__ ssum, float* __restrict__ out, int Nt) {
    int idx = blockIdx.x * blockDim.x + threadIdx.x;
    if (idx >= Nt * DD) return;
    int t = idx >> 7, c = idx & 127, h = c >> 4;
    float denom = ssum[(size_t)t * NHEAD + h] + 1e-16f;
    out[idx] = 0.5f * s[idx] + 0.5f * (num[idx] / denom);
}

// ---------------------------------------------------------------- BatchNorm stats (per-column sum, sumsq)
__global__ void bn_stats(const float* __restrict__ h, float* __restrict__ sums, int N) {
    int col = threadIdx.x;                  // 128 threads
    int r0 = blockIdx.x * 256;
    int r1 = r0 + 256; if (r1 > N) r1 = N;
    float s = 0.0f, s2 = 0.0f;
    for (int r = r0; r < r1; ++r) {
        float x = h[(size_t)r * DD + col];
        s += x; s2 += x * x;
    }
    atomicAdd(&sums[col], s);
    atomicAdd(&sums[DD + col], s2);
}

__global__ void bn_apply(float* __restrict__ h, const float* __restrict__ sums,
                         const float* __restrict__ g, const float* __restrict__ b, int N) {
    int idx = blockIdx.x * blockDim.x + threadIdx.x;
    if (idx >= N * DD) return;
    int col = idx & 127;
    float invN = 1.0f / (float)N;
    float mu = sums[col] * invN;
    float var = sums[DD + col] * invN - mu * mu;
    h[idx] = (h[idx] - mu) * rsqrtf(var + 1e-5f) * g[col] + b[col];
}

// ================================================================ launch
extern "C" void kernel_launch(void* const* d_in, const int* in_sizes, int n_in,
                              void* d_out, int out_size, void* d_ws, size_t ws_size,
                              hipStream_t stream) {
    const float* x    = (const float*)d_in[0];
    const float* ttab = (const float*)d_in[1];
    const float* atab = (const float*)d_in[2];
    const float* Wq0 = (const float*)d_in[3];
    const float* Wk0 = (const float*)d_in[4];
    const float* Wv0 = (const float*)d_in[5];
    const float* Ws0 = (const float*)d_in[6];
    const float* Wq1 = (const float*)d_in[7];
    const float* Wk1 = (const float*)d_in[8];
    const float* Wv1 = (const float*)d_in[9];
    const float* Ws1 = (const float*)d_in[10];
    const float* Wek1 = (const float*)d_in[11];
    const float* Wev1 = (const float*)d_in[12];
    const float* Wq2 = (const float*)d_in[13];
    const float* Wk2 = (const float*)d_in[14];
    const float* Wv2 = (const float*)d_in[15];
    const float* Ws2 = (const float*)d_in[16];
    const float* g0  = (const float*)d_in[17];
    const float* b0  = (const float*)d_in[18];
    const int* src0 = (const int*)d_in[19];
    const int* dst0 = (const int*)d_in[20];
    const int* src1 = (const int*)d_in[21];
    const int* dst1 = (const int*)d_in[22];
    const int* src2 = (const int*)d_in[23];
    const int* dst2 = (const int*)d_in[24];
    const int* etyp = (const int*)d_in[25];
    (void)Wq0; (void)Ws0; (void)in_sizes; (void)n_in; (void)out_size; (void)ws_size;

    float* ws = (float*)d_ws;
    size_t o = 0;
    float* K   = ws + o; o += (size_t)NXR * DD;     // k proj (max 150000 rows), reused by layers 1/2
    float* V   = ws + o; o += (size_t)NXR * DD;     // v proj
    float* Q   = ws + o; o += (size_t)N1R * DD;     // q proj (max 50000 rows)
    float* S   = ws + o; o += (size_t)N1R * DD;     // skip proj
    float* LG  = ws + o; o += (size_t)E0N * NHEAD;  // per-edge logits (max E0)
    float* M   = ws + o; o += (size_t)N1R * NHEAD;  // segment max
    float* SS  = ws + o; o += (size_t)N1R * NHEAD;  // segment sum
    float* NUM = ws + o; o += (size_t)N1R * DD;     // segment numerator
    float* H1  = ws + o; o += (size_t)N1R * DD;     // layer-0 output
    float* H2  = ws + o; o += (size_t)N2R * DD;     // layer-1 output
    float* TK  = ws + o; o += (size_t)NTYPE * DD;   // edge-type K table
    float* TV  = ws + o; o += (size_t)NTYPE * DD;   // edge-type V table
    float* BN  = ws + o; o += 2 * DD;               // bn sum / sumsq

    auto fill = [&](float* p, float v, int n) {
        fill_f32<<<(n + 255) / 256, 256, 0, stream>>>(p, v, n);
    };
    auto gemm = [&](const float* X, const float* W, float* Y, int R) {
        gemm128_wmma<<<(R + 127) / 128, 256, 0, stream>>>(X, W, Y, R);
    };
    // edge kernels: one wave32 per edge, 8 edges per 256-thread block
    auto eblocks = [](int E) { return (E + 7) / 8; };

    // ---------------- Layer 0: x_tgt = zero rows => q0 = s0 = 0
    fill(Q, 0.0f, N1R * DD);
    fill(S, 0.0f, N1R * DD);
    fill(M, -INFINITY, N1R * NHEAD);
    fill(SS, 0.0f, N1R * NHEAD);
    fill(NUM, 0.0f, N1R * DD);
    gemm(x, Wk0, K, NXR);
    gemm(x, Wv0, V, NXR);
    edge_logits<<<eblocks(E0N), 256, 0, stream>>>(
        Q, K, src0, dst0, nullptr, nullptr, LG, M, E0N, SPLITN);
    edge_accum<<<eblocks(E0N), 256, 0, stream>>>(
        LG, M, V, src0, dst0, nullptr, nullptr, SS, NUM, E0N, SPLITN);
    finalize_k<<<(N1R * DD + 255) / 256, 256, 0, stream>>>(S, NUM, SS, H1, N1R);
    fill(BN, 0.0f, 2 * DD);
    bn_stats<<<(N1R + 255) / 256, 128, 0, stream>>>(H1, BN, N1R);
    bn_apply<<<(N1R * DD + 255) / 256, 256, 0, stream>>>(H1, BN, g0, b0, N1R);

    // ---------------- Layer 1 (edge-type embeddings added to K/V)
    fill(M, -INFINITY, N2R * NHEAD);
    fill(SS, 0.0f, N2R * NHEAD);
    fill(NUM, 0.0f, N2R * DD);
    gemm(ttab, Wek1, TK, NTYPE);
    gemm(atab, Wev1, TV, NTYPE);
    gemm(H1, Wq1, Q, N2R);
    gemm(H1, Wk1, K, N1R);
    gemm(H1, Wv1, V, N1R);
    gemm(H1, Ws1, S, N2R);
    edge_logits<<<eblocks(E1N), 256, 0, stream>>>(
        Q, K, src1, dst1, TK, etyp, LG, M, E1N, 0);
    edge_accum<<<eblocks(E1N), 256, 0, stream>>>(
        LG, M, V, src1, dst1, TV, etyp, SS, NUM, E1N, 0);
    finalize_k<<<(N2R * DD + 255) / 256, 256, 0, stream>>>(S, NUM, SS, H2, N2R);

    // ---------------- Layer 2
    fill(M, -INFINITY, N3R * NHEAD);
    fill(SS, 0.0f, N3R * NHEAD);
    fill(NUM, 0.0f, N3R * DD);
    gemm(H2, Wq2, Q, N3R);
    gemm(H2, Wk2, K, N2R);
    gemm(H2, Wv2, V, N2R);
    gemm(H2, Ws2, S, N3R);
    edge_logits<<<eblocks(E2N), 256, 0, stream>>>(
        Q, K, src2, dst2, nullptr, nullptr, LG, M, E2N, 0);
    edge_accum<<<eblocks(E2N), 256, 0, stream>>>(
        LG, M, V, src2, dst2, nullptr, nullptr, SS, NUM, E2N, 0);
    finalize_k<<<(N3R * DD + 255) / 256, 256, 0, stream>>>(S, NUM, SS, (float*)d_out, N3R);
}